// GaussianRasterizer_90890097918473
// MI455X (gfx1250) — compile-verified
//
#include <hip/hip_runtime.h>
#include <hip/hip_bf16.h>
#include <math.h>

#define IMG_H 128
#define IMG_W 128
#define N_PIX (IMG_H * IMG_W)
#define TANFOVX 0.5f
#define TANFOVY 0.5f
#define N_GAUSS 2048
#define MIN_ALPHA (1.0f / 255.0f)
#define LOWPASS 0.3f
#define REC_STRIDE 12   // floats per sorted gaussian record (48 B)

typedef __attribute__((ext_vector_type(2))) float v2f;
typedef __attribute__((ext_vector_type(8))) float v8f;

// ---------------------------------------------------------------------------
// Kernel 1: per-gaussian preprocess -> prep record, sort key, radii output
// prep layout (12 floats): mx,my,cA,cB,cC,op,cr,cg,cb,invd,validFlag,pad
// ---------------------------------------------------------------------------
__global__ __launch_bounds__(256) void gs_prep_kernel(
    const float* __restrict__ means3D, const float* __restrict__ opac,
    const float* __restrict__ scales, const float* __restrict__ rots,
    const float* __restrict__ cols, const float* __restrict__ viewM,
    const float* __restrict__ projM, float* __restrict__ prep,
    float* __restrict__ keys, int* __restrict__ radii_out) {
  int g = blockIdx.x * blockDim.x + threadIdx.x;
  if (g >= N_GAUSS) return;

  float m0 = means3D[3 * g + 0], m1 = means3D[3 * g + 1], m2 = means3D[3 * g + 2];

  // Quaternion -> rotation
  float q0 = rots[4 * g + 0], q1 = rots[4 * g + 1], q2 = rots[4 * g + 2], q3 = rots[4 * g + 3];
  float qn = rsqrtf(q0 * q0 + q1 * q1 + q2 * q2 + q3 * q3);
  float r = q0 * qn, x = q1 * qn, y = q2 * qn, z = q3 * qn;
  float R00 = 1.f - 2.f * (y * y + z * z), R01 = 2.f * (x * y - r * z), R02 = 2.f * (x * z + r * y);
  float R10 = 2.f * (x * y + r * z), R11 = 1.f - 2.f * (x * x + z * z), R12 = 2.f * (y * z - r * x);
  float R20 = 2.f * (x * z - r * y), R21 = 2.f * (y * z + r * x), R22 = 1.f - 2.f * (x * x + y * y);

  float s0 = scales[3 * g + 0], s1 = scales[3 * g + 1], s2 = scales[3 * g + 2];
  float M00 = R00 * s0, M01 = R01 * s1, M02 = R02 * s2;
  float M10 = R10 * s0, M11 = R11 * s1, M12 = R12 * s2;
  float M20 = R20 * s0, M21 = R21 * s1, M22 = R22 * s2;
  // cov3D = M M^T (symmetric)
  float C00 = M00 * M00 + M01 * M01 + M02 * M02;
  float C01 = M00 * M10 + M01 * M11 + M02 * M12;
  float C02 = M00 * M20 + M01 * M21 + M02 * M22;
  float C11 = M10 * M10 + M11 * M11 + M12 * M12;
  float C12 = M10 * M20 + M11 * M21 + M12 * M22;
  float C22 = M20 * M20 + M21 * M21 + M22 * M22;

  // view transform (row-major 4x4)
  float t0 = viewM[0] * m0 + viewM[1] * m1 + viewM[2] * m2 + viewM[3];
  float t1 = viewM[4] * m0 + viewM[5] * m1 + viewM[6] * m2 + viewM[7];
  float tz = viewM[8] * m0 + viewM[9] * m1 + viewM[10] * m2 + viewM[11];
  float tzs = (fabsf(tz) > 1e-6f) ? tz : 1e-6f;

  float limx = 1.3f * TANFOVX, limy = 1.3f * TANFOVY;
  float txc = fminf(fmaxf(t0 / tzs, -limx), limx) * tzs;
  float tyc = fminf(fmaxf(t1 / tzs, -limy), limy) * tzs;

  float fx = IMG_W / (2.f * TANFOVX), fy = IMG_H / (2.f * TANFOVY);
  float z2 = tzs * tzs;
  float J00 = fx / tzs, J02 = -fx * txc / z2;
  float J11 = fy / tzs, J12 = -fy * tyc / z2;
  // T = J @ V[:3,:3]
  float T00 = J00 * viewM[0] + J02 * viewM[8];
  float T01 = J00 * viewM[1] + J02 * viewM[9];
  float T02 = J00 * viewM[2] + J02 * viewM[10];
  float T10 = J11 * viewM[4] + J12 * viewM[8];
  float T11 = J11 * viewM[5] + J12 * viewM[9];
  float T12 = J11 * viewM[6] + J12 * viewM[10];
  // U = T @ C
  float U00 = T00 * C00 + T01 * C01 + T02 * C02;
  float U01 = T00 * C01 + T01 * C11 + T02 * C12;
  float U02 = T00 * C02 + T01 * C12 + T02 * C22;
  float U10 = T10 * C00 + T11 * C01 + T12 * C02;
  float U11 = T10 * C01 + T11 * C11 + T12 * C12;
  float U12 = T10 * C02 + T11 * C12 + T12 * C22;
  float cov00 = U00 * T00 + U01 * T01 + U02 * T02;
  float cov01 = U00 * T10 + U01 * T11 + U02 * T12;
  float cov11 = U10 * T10 + U11 * T11 + U12 * T12;

  float a = cov00 + LOWPASS, cc = cov11 + LOWPASS, b = cov01;
  float det = a * cc - b * b;
  float det_s = (det != 0.f) ? det : 1.f;
  float conA = cc / det_s, conB = -b / det_s, conC = a / det_s;
  float mid = 0.5f * (a + cc);
  float lam = mid + sqrtf(fmaxf(0.1f, mid * mid - det));
  float radii = ceilf(3.f * sqrtf(lam));

  // projection
  float p0 = projM[0] * m0 + projM[1] * m1 + projM[2] * m2 + projM[3];
  float p1 = projM[4] * m0 + projM[5] * m1 + projM[6] * m2 + projM[7];
  float p3 = projM[12] * m0 + projM[13] * m1 + projM[14] * m2 + projM[15];
  float invw = 1.f / (p3 + 1e-7f);
  float px = ((p0 * invw + 1.f) * IMG_W - 1.f) * 0.5f;
  float py = ((p1 * invw + 1.f) * IMG_H - 1.f) * 0.5f;

  bool valid = (tz > 0.2f) && (det > 0.f) && (radii > 0.f);
  keys[g] = valid ? tz : __builtin_inff();

  float* pr = prep + REC_STRIDE * g;
  pr[0] = px; pr[1] = py; pr[2] = conA; pr[3] = conB; pr[4] = conC;
  pr[5] = opac[g];
  pr[6] = cols[3 * g + 0]; pr[7] = cols[3 * g + 1]; pr[8] = cols[3 * g + 2];
  pr[9] = 1.f / (valid ? tz : 1.f);
  pr[10] = valid ? 1.f : 0.f;
  pr[11] = 0.f;

  radii_out[g] = valid ? (int)radii : 0;
}

// ---------------------------------------------------------------------------
// Kernel 2: one-block bitonic sort of (key, index), N=2048, 1024 threads
// ---------------------------------------------------------------------------
__global__ __launch_bounds__(1024) void gs_sort_kernel(
    const float* __restrict__ keys, int* __restrict__ idx_out) {
  __shared__ float sk[N_GAUSS];
  __shared__ int si[N_GAUSS];
  int t = threadIdx.x;
  for (int i = t; i < N_GAUSS; i += 1024) { sk[i] = keys[i]; si[i] = i; }
  __syncthreads();
  for (int k = 2; k <= N_GAUSS; k <<= 1) {
    for (int j = k >> 1; j > 0; j >>= 1) {
      for (int base = 0; base < N_GAUSS; base += 1024) {
        int i = base + t;
        int ixj = i ^ j;
        if (ixj > i) {
          bool up = ((i & k) == 0);
          float ka = sk[i], kb = sk[ixj];
          bool sw = up ? (ka > kb) : (ka < kb);
          if (sw) {
            sk[i] = kb; sk[ixj] = ka;
            int tmp = si[i]; si[i] = si[ixj]; si[ixj] = tmp;
          }
        }
      }
      __syncthreads();
    }
  }
  for (int i = t; i < N_GAUSS; i += 1024) idx_out[i] = si[i];
}

// ---------------------------------------------------------------------------
// Kernel 3: gather into sorted records with WMMA-ready bilinear features
// rec (12 floats): f0..f5, op, cr, cg, cb, invd, pad
// power(g,p) = f0*1 + f1*x + f2*y + f3*x^2 + f4*y^2 + f5*x*y
// ---------------------------------------------------------------------------
__global__ __launch_bounds__(256) void gs_gather_kernel(
    const float* __restrict__ prep, const int* __restrict__ idx,
    float* __restrict__ rec) {
  int s = blockIdx.x * blockDim.x + threadIdx.x;
  if (s >= N_GAUSS) return;
  int g = idx[s];
  const float* pr = prep + REC_STRIDE * g;
  float mx = pr[0], my = pr[1], cA = pr[2], cB = pr[3], cC = pr[4];
  float op = pr[5], cr = pr[6], cg = pr[7], cb = pr[8], iv = pr[9];
  bool valid = pr[10] != 0.f;
  float f0 = -0.5f * (cA * mx * mx + cC * my * my) - cB * mx * my;
  float f1 = cA * mx + cB * my;
  float f2 = cC * my + cB * mx;
  float f3 = -0.5f * cA;
  float f4 = -0.5f * cC;
  float f5 = -cB;
  if (!valid) { f0 = f1 = f2 = f3 = f4 = f5 = 0.f; op = 0.f; }
  float* rc = rec + REC_STRIDE * s;
  rc[0] = f0; rc[1] = f1; rc[2] = f2; rc[3] = f3; rc[4] = f4; rc[5] = f5;
  rc[6] = op; rc[7] = cr; rc[8] = cg; rc[9] = cb; rc[10] = iv; rc[11] = 0.f;
}

// ---------------------------------------------------------------------------
// Kernel 4: rasterize. One wave32 handles 16 pixels (two lanes per pixel).
//
// Stage 0: the whole sorted record array (96 KB) is copied global->LDS once
// per workgroup with CDNA5 async-to-LDS b128 ops (ASYNCcnt), so the 128-chunk
// blend loop runs entirely out of LDS (no redundant global traffic; each
// record is read once from HBM/L2 per block instead of once per lane).
//
// Per 16-gaussian chunk: power tile via two chained V_WMMA_F32_16X16X4_F32
// (M=gaussian, N=pixel, K=6 padded to 8), then serial alpha-blend.
// D layout: lane L<16 -> pixel L, gaussians 0-7 in 8 VGPRs; lane L+16 ->
// pixel L, gaussians 8-15. Halves stitched via shuffles.
// ---------------------------------------------------------------------------
__global__ __launch_bounds__(256) void gs_raster_kernel(
    const float* __restrict__ rec, const float* __restrict__ bg,
    float* __restrict__ outColor, float* __restrict__ outInv) {
  __shared__ float srec[N_GAUSS * REC_STRIDE];  // 98304 B of LDS

  // ---- async global -> LDS staging (b128 = 4 floats per op) ----
  {
    unsigned lds_base = (unsigned)(size_t)(void*)&srec[0];
    const int n128 = (N_GAUSS * REC_STRIDE) / 4;  // 6144 16-byte packets
    for (int i = threadIdx.x; i < n128; i += 256) {
      unsigned loff = lds_base + (unsigned)i * 16u;
      const float* gp = rec + i * 4;
      asm volatile("global_load_async_to_lds_b128 %0, %1, off"
                   :: "v"(loff), "v"(gp)
                   : "memory");
    }
    asm volatile("s_wait_asynccnt 0x0" ::: "memory");
  }
  __syncthreads();

  int tid = blockIdx.x * blockDim.x + threadIdx.x;
  int lane = threadIdx.x & 31;
  int wave = tid >> 5;          // 1024 waves total, 16 pixels each
  int lid = lane & 15;
  int hf = lane >> 4;           // 0 = gaussians 0-7, 1 = gaussians 8-15
  int p = wave * 16 + lid;      // flat pixel, 16 consecutive in a row
  float xf = (float)(p & (IMG_W - 1));
  float yf = (float)(p >> 7);

  // B matrix per lane (pixel features): low half carries K0,K1; high K2,K3.
  v2f b1, b2;
  b1.x = hf ? yf : 1.f;
  b1.y = hf ? (xf * xf) : xf;
  b2.x = hf ? 0.f : (yf * yf);
  b2.y = hf ? 0.f : (xf * yf);

  float Tpix = 1.f;
  float accR = 0.f, accG = 0.f, accB = 0.f, accI = 0.f;

  for (int chunk = 0; chunk < N_GAUSS / 16; ++chunk) {
    int cb = chunk * 16 * REC_STRIDE;

    // A matrix per lane (gaussian features): low half K0,K1; high half K2,K3.
    v2f lo01 = *(const v2f*)&srec[cb + lid * REC_STRIDE + 0];
    v2f lo23 = *(const v2f*)&srec[cb + lid * REC_STRIDE + 2];
    v2f lo45 = *(const v2f*)&srec[cb + lid * REC_STRIDE + 4];
    v2f a1, a2;
    a1.x = hf ? lo23.x : lo01.x;
    a1.y = hf ? lo23.y : lo01.y;
    a2.x = hf ? 0.f : lo45.x;
    a2.y = hf ? 0.f : lo45.y;

    v8f c = {};
    c = __builtin_amdgcn_wmma_f32_16x16x4_f32(false, a1, false, b1, (short)0, c,
                                              false, false);
    c = __builtin_amdgcn_wmma_f32_16x16x4_f32(false, a2, false, b2, (short)0, c,
                                              false, false);

    // Serial blend over this lane's 8 gaussians (m = hf*8 + r2), sorted order.
    int prm = cb + hf * 8 * REC_STRIDE;
    float Tl = 1.f, sr = 0.f, sg = 0.f, sb = 0.f, si = 0.f;
#pragma unroll
    for (int r2 = 0; r2 < 8; ++r2) {
      int q = prm + r2 * REC_STRIDE;
      v2f opr = *(const v2f*)&srec[q + 6];   // op, colR
      v2f cgb = *(const v2f*)&srec[q + 8];   // colG, colB
      float qi = srec[q + 10];               // invdepth
      float pw = c[r2];
      float al = fminf(0.99f, opr.x * __expf(pw));
      if (pw > 0.f || al < MIN_ALPHA) al = 0.f;
      float w = al * Tl;
      sr += w * opr.y; sg += w * cgb.x; sb += w * cgb.y; si += w * qi;
      Tl *= 1.f - al;
    }

    // Stitch halves: high half scaled by low half's transmittance.
    float Tlow = __shfl(Tl, lid, 32);
    float Thigh = __shfl(Tl, lid + 16, 32);
    float scale = Tpix * (hf ? Tlow : 1.f);
    accR += scale * sr; accG += scale * sg; accB += scale * sb; accI += scale * si;
    Tpix *= Tlow * Thigh;
  }

  // Fold the two half-lane accumulators per pixel.
  accR += __shfl_xor(accR, 16, 32);
  accG += __shfl_xor(accG, 16, 32);
  accB += __shfl_xor(accB, 16, 32);
  accI += __shfl_xor(accI, 16, 32);

  if (hf == 0) {
    float b0 = bg[0], b1s = bg[1], b2s = bg[2];
    outColor[0 * N_PIX + p] = accR + Tpix * b0;
    outColor[1 * N_PIX + p] = accG + Tpix * b1s;
    outColor[2 * N_PIX + p] = accB + Tpix * b2s;
    outInv[p] = accI;
  }
}

// ---------------------------------------------------------------------------
extern "C" void kernel_launch(void* const* d_in, const int* in_sizes, int n_in,
                              void* d_out, int out_size, void* d_ws, size_t ws_size,
                              hipStream_t stream) {
  const float* means3D = (const float*)d_in[0];
  // d_in[1] = means2D (unused)
  const float* opacities = (const float*)d_in[2];
  const float* scales = (const float*)d_in[3];
  const float* rotations = (const float*)d_in[4];
  const float* colors = (const float*)d_in[5];
  const float* bg = (const float*)d_in[6];
  const float* viewmatrix = (const float*)d_in[7];
  const float* projmatrix = (const float*)d_in[8];

  // Workspace layout (floats): prep[2048*12] | keys[2048] | idx[2048] | rec[2048*12]
  float* prep = (float*)d_ws;
  float* keys = prep + N_GAUSS * REC_STRIDE;
  int* idx = (int*)(keys + N_GAUSS);
  float* rec = (float*)(idx + N_GAUSS);

  // Output: color[3*128*128] | radii[2048] (int32 bits) | invdepth[128*128]
  float* outColor = (float*)d_out;
  int* outRadii = (int*)d_out + 3 * N_PIX;
  float* outInv = (float*)d_out + 3 * N_PIX + N_GAUSS;

  gs_prep_kernel<<<N_GAUSS / 256, 256, 0, stream>>>(
      means3D, opacities, scales, rotations, colors, viewmatrix, projmatrix,
      prep, keys, outRadii);
  gs_sort_kernel<<<1, 1024, 0, stream>>>(keys, idx);
  gs_gather_kernel<<<N_GAUSS / 256, 256, 0, stream>>>(prep, idx, rec);
  gs_raster_kernel<<<(N_PIX * 2) / 256, 256, 0, stream>>>(rec, bg, outColor, outInv);
}